// GCN_90915867721778
// MI455X (gfx1250) — compile-verified
//
#include <hip/hip_runtime.h>
#include <hip/hip_bf16.h>

typedef __attribute__((ext_vector_type(2))) float v2f;
typedef __attribute__((ext_vector_type(8))) float v8f;

// ---------------------------------------------------------------------------
// Elementwise / scatter helpers
// ---------------------------------------------------------------------------
__global__ void zero_f32_kernel(float* __restrict__ p, long long n) {
  long long i = (long long)blockIdx.x * blockDim.x + threadIdx.x;
  if (i < n) p[i] = 0.0f;
}

__global__ void degree_accum_kernel(const long long* __restrict__ ei,
                                    float* __restrict__ deg, long long E) {
  long long e = (long long)blockIdx.x * blockDim.x + threadIdx.x;
  if (e < E) {
    long long d = ei[E + e];               // dst row of edge_index
    unsafeAtomicAdd(deg + d, 1.0f);        // global_atomic_add_f32 (no return)
  }
}

__global__ void make_dinv_kernel(float* __restrict__ deg_dinv, long long N) {
  long long i = (long long)blockIdx.x * blockDim.x + threadIdx.x;
  if (i < N) deg_dinv[i] = rsqrtf(deg_dinv[i] + 1.0f);   // deg>=0 -> safe
}

__global__ void edge_norm_kernel(const long long* __restrict__ ei,
                                 const float* __restrict__ dinv,
                                 float* __restrict__ nrm, long long E) {
  long long e = (long long)blockIdx.x * blockDim.x + threadIdx.x;
  if (e < E) nrm[e] = dinv[ei[e]] * dinv[ei[E + e]];
}

// One edge handled by 16 threads, 4 channels (float4) each -> the 16 threads
// collectively stream one contiguous 256B feature row (good coalescing).
__global__ void edge_agg_kernel(const long long* __restrict__ ei,
                                const float* __restrict__ nrm,
                                const float* __restrict__ h,
                                float* __restrict__ agg, long long E) {
  long long idx = (long long)blockIdx.x * blockDim.x + threadIdx.x;
  if (idx >= E * 16) return;
  long long e = idx >> 4;
  int c = ((int)idx & 15) << 2;
  long long s = ei[e];
  long long d = ei[E + e];
  float w = nrm[e];
  const float4 hv = *reinterpret_cast<const float4*>(h + s * 64 + c);
  float* ap = agg + d * 64 + c;
  unsafeAtomicAdd(ap + 0, hv.x * w);
  unsafeAtomicAdd(ap + 1, hv.y * w);
  unsafeAtomicAdd(ap + 2, hv.z * w);
  unsafeAtomicAdd(ap + 3, hv.w * w);
}

// out = (agg + hself*dinv^2 + bias), optional ReLU.  64 channels per node.
__global__ void post_kernel(const float* __restrict__ agg,
                            const float* __restrict__ hself,
                            const float* __restrict__ dinv,
                            const float* __restrict__ bias,
                            float* __restrict__ out, int do_relu, long long N) {
  long long idx = (long long)blockIdx.x * blockDim.x + threadIdx.x;
  if (idx >= N * 64) return;
  long long n = idx >> 6;
  int c = (int)idx & 63;
  float di = dinv[n];
  float v = agg[idx] + hself[idx] * (di * di) + bias[c];
  if (do_relu) v = fmaxf(v, 0.0f);
  out[idx] = v;
}

// ---------------------------------------------------------------------------
// WMMA GEMM:  out[M x 64] = A[M x K] * W[K x 64]   (fp32, v_wmma_f32_16x16x4_f32)
// Block = 256 threads = 8 wave32s; each wave owns one 16-row strip and the
// full 64-wide output (4 accumulator tiles of v8f).
//
// W is staged in LDS with a k-pair interleave:
//   sW[(k>>1)*128 + n*2 + (k&1)] = W[k*64 + n]
// so each lane's B fragment { W[k+kp][n], W[k+kp+1][n] } is one aligned
// ds_load_b64 directly into the even-aligned v2f WMMA operand (no VGPR
// repacking movs, 4 DS loads per k-step instead of 8).
//
// A-fragment layout (ISA 7.12.2, 32-bit A 16x4): lanes 0-15 -> K=k,k+1;
// lanes 16-31 -> K=k+2,k+3; row M = lane&15.  B mirrors with N = lane&15.
// ---------------------------------------------------------------------------
template <int K>
__global__ __launch_bounds__(256) void gemm_wmma_kernel(
    const float* __restrict__ A, const float* __restrict__ W,
    float* __restrict__ out, int M) {
  __shared__ float sW[K * 64];
  for (int i = threadIdx.x; i < K * 64; i += blockDim.x) {
    int k = i >> 6;
    int n = i & 63;
    sW[(k >> 1) * 128 + n * 2 + (k & 1)] = W[i];
  }
  __syncthreads();

  const int wave = threadIdx.x >> 5;
  const int lane = threadIdx.x & 31;
  const int rowBlock = blockIdx.x * 8 + wave;
  if (rowBlock * 16 >= M) return;              // wave-uniform: EXEC stays all-1s

  const int row0 = rowBlock * 16;
  const int m  = lane & 15;
  const int n  = lane & 15;
  const int kp = (lane >> 4) * 2;              // 0 for lanes 0-15, 2 for 16-31

  v8f c0 = {}, c1 = {}, c2 = {}, c3 = {};
  const float* arow = A + (size_t)(row0 + m) * K + kp;

#pragma unroll
  for (int k = 0; k < K; k += 4) {
    const float2 av = *reinterpret_cast<const float2*>(arow + k);  // 8B aligned
    v2f a; a[0] = av.x; a[1] = av.y;

    // interleaved B rows for this k-step (kp keeps it lane-correct)
    const float* wp = sW + ((k + kp) >> 1) * 128;
    const v2f b0 = *reinterpret_cast<const v2f*>(wp + (n)      * 2);
    const v2f b1 = *reinterpret_cast<const v2f*>(wp + (16 + n) * 2);
    const v2f b2 = *reinterpret_cast<const v2f*>(wp + (32 + n) * 2);
    const v2f b3 = *reinterpret_cast<const v2f*>(wp + (48 + n) * 2);

    c0 = __builtin_amdgcn_wmma_f32_16x16x4_f32(false, a, false, b0, (short)0, c0, false, false);
    c1 = __builtin_amdgcn_wmma_f32_16x16x4_f32(false, a, false, b1, (short)0, c1, false, false);
    c2 = __builtin_amdgcn_wmma_f32_16x16x4_f32(false, a, false, b2, (short)0, c2, false, false);
    c3 = __builtin_amdgcn_wmma_f32_16x16x4_f32(false, a, false, b3, (short)0, c3, false, false);
  }

  // C/D layout: lanes 0-15: VGPR j -> (M=j, N=lane); lanes 16-31: (M=j+8, N=lane-16)
  const int mhalf = (lane >> 4) * 8;
  float* orow = out + (size_t)row0 * 64;
#pragma unroll
  for (int j = 0; j < 8; ++j) {
    const size_t r = (size_t)(mhalf + j) * 64;
    orow[r + n]      = c0[j];
    orow[r + 16 + n] = c1[j];
    orow[r + 32 + n] = c2[j];
    orow[r + 48 + n] = c3[j];
  }
}

// ---------------------------------------------------------------------------
static inline unsigned cdiv(long long a, long long b) { return (unsigned)((a + b - 1) / b); }

extern "C" void kernel_launch(void* const* d_in, const int* in_sizes, int n_in,
                              void* d_out, int out_size, void* d_ws, size_t ws_size,
                              hipStream_t stream) {
  const float*     x  = (const float*)d_in[0];
  const long long* ei = (const long long*)d_in[1];     // int64 (2, E)
  const float*     W1 = (const float*)d_in[2];
  const float*     b1 = (const float*)d_in[3];
  const float*     W2 = (const float*)d_in[4];
  const float*     b2 = (const float*)d_in[5];
  float*           out = (float*)d_out;

  const long long N = in_sizes[0] / 128;   // 100000
  const long long E = in_sizes[1] / 2;     // 1600000

  // workspace layout (all 16B-aligned for these sizes)
  float* dinv = (float*)d_ws;              // N floats
  float* nrm  = dinv + N;                  // E floats
  float* bufA = nrm + E;                   // N*64 floats
  float* bufB = bufA + N * 64;             // N*64 floats

  const int T = 256;
  const unsigned gemmBlocks = cdiv(cdiv(N, 16), 8);

  // --- normalization terms (shared by both layers) ---
  zero_f32_kernel<<<cdiv(N, T), T, 0, stream>>>(dinv, N);
  degree_accum_kernel<<<cdiv(E, T), T, 0, stream>>>(ei, dinv, E);
  make_dinv_kernel<<<cdiv(N, T), T, 0, stream>>>(dinv, N);
  edge_norm_kernel<<<cdiv(E, T), T, 0, stream>>>(ei, dinv, nrm, E);

  // --- layer 1: h1 = relu(gcn(x, W1, b1)) ---
  gemm_wmma_kernel<128><<<gemmBlocks, T, 0, stream>>>(x, W1, bufA, (int)N);
  zero_f32_kernel<<<cdiv(N * 64, T), T, 0, stream>>>(bufB, N * 64);
  edge_agg_kernel<<<cdiv(E * 16, T), T, 0, stream>>>(ei, nrm, bufA, bufB, E);
  post_kernel<<<cdiv(N * 64, T), T, 0, stream>>>(bufB, bufA, dinv, b1, bufA, 1, N);

  // --- layer 2: out = gcn(h1, W2, b2) ---
  gemm_wmma_kernel<64><<<gemmBlocks, T, 0, stream>>>(bufA, W2, bufB, (int)N);
  zero_f32_kernel<<<cdiv(N * 64, T), T, 0, stream>>>(out, N * 64);
  edge_agg_kernel<<<cdiv(E * 16, T), T, 0, stream>>>(ei, nrm, bufB, out, E);
  post_kernel<<<cdiv(N * 64, T), T, 0, stream>>>(out, bufB, dinv, b2, out, 0, N);
}